// EnhancedLossModule_63170378989858
// MI455X (gfx1250) — compile-verified
//
#include <hip/hip_runtime.h>
#include <math.h>

// ---------------------------------------------------------------------------
// EnhancedLossModule for MI455X (gfx1250, wave32)
// Inputs : pred [2048,1000] f32, target [2048] i32, features [2048,512] f32
// Output : 5 f32 = {contrastive, triplet, focal, label_smooth, total}
// Strategy: fused fp32-WMMA Gram kernel (contrastive + distance rows in LDS
//           + triplet scan per 16-row block), softmax kernel for focal/ls,
//           deterministic tree reductions (no atomics).
// ---------------------------------------------------------------------------

typedef __attribute__((ext_vector_type(2))) float v2f;
typedef __attribute__((ext_vector_type(8))) float v8f;

#define B_N      2048
#define D_K      512
#define C_CLS    1000
#define ROWS     16          // rows of G per workgroup
#define NWAVES   8           // 256 threads
#define NTILES   (B_N / 16)  // 128 column tiles of width 16
#define INV_T    (1.0f / 0.07f)
#define C_MARGIN 0.5f
#define T_MARGIN 1.0f

// padded LDS strides (floats) to avoid bank conflicts
#define DS_STRIDE 2064       // 2048 + 16

// dynamic-LDS byte offsets
#define OFF_DS    0
#define OFF_LBL   (ROWS * DS_STRIDE * 4)            // 132096
#define OFF_RN    (OFF_LBL + B_N * 4)               // 140288
#define OFF_INVN  (OFF_RN + B_N * 4)                // 148480
#define OFF_RED   (OFF_INVN + B_N * 4)              // 156672
#define DYN_LDS   (OFF_RED + 256 * 4)               // 157696 bytes < 320KB

// workspace float layout
#define WS_R      0
#define WS_INVN   (WS_R + B_N)
#define WS_FOCAL  (WS_INVN + B_N)
#define WS_LSM    (WS_FOCAL + B_N)
#define WS_CPART  (WS_LSM + B_N)
#define WS_TPART  (WS_CPART + 128)

__device__ __forceinline__ float block_reduce_sum(float v, float* red, int tid) {
    red[tid] = v;
    __syncthreads();
    for (int s = 128; s > 0; s >>= 1) {
        if (tid < s) red[tid] += red[tid + s];
        __syncthreads();
    }
    float r = red[0];
    __syncthreads();
    return r;
}

__device__ __forceinline__ float block_reduce_max(float v, float* red, int tid) {
    red[tid] = v;
    __syncthreads();
    for (int s = 128; s > 0; s >>= 1) {
        if (tid < s) red[tid] = fmaxf(red[tid], red[tid + s]);
        __syncthreads();
    }
    float r = red[0];
    __syncthreads();
    return r;
}

// -------------------------------------------------------------------- norms
__global__ void el_norms_kernel(const float* __restrict__ F, float* __restrict__ ws) {
    const int tid  = threadIdx.x;
    const int lane = tid & 31;
    const int wave = tid >> 5;
    const int row  = blockIdx.x * 8 + wave;
    const float* fr = F + (size_t)row * D_K;
    float s = 0.0f;
#pragma unroll
    for (int t = 0; t < D_K / 32; ++t) {
        float x = fr[lane + 32 * t];
        s = fmaf(x, x, s);
    }
#pragma unroll
    for (int off = 16; off >= 1; off >>= 1) s += __shfl_xor(s, off, 32);
    if (lane == 0) {
        ws[WS_R + row]    = s;
        ws[WS_INVN + row] = 1.0f / sqrtf(s);
    }
}

// ------------------------------------------------ fused gram / contrastive /
// triplet kernel: one workgroup = 16 rows of G over all 2048 columns.
__global__ void el_gram_fused_kernel(const float* __restrict__ F,
                                     const int* __restrict__ labels,
                                     const float* __restrict__ ws_in,
                                     float* __restrict__ ws_out) {
    extern __shared__ char smem[];
    float* dsm  = (float*)(smem + OFF_DS);    // [16][DS_STRIDE] distance rows
    int*   lbl  = (int*)  (smem + OFF_LBL);   // [2048]
    float* rn   = (float*)(smem + OFF_RN);    // [2048] squared norms
    float* invn = (float*)(smem + OFF_INVN);  // [2048] 1/norm
    float* red  = (float*)(smem + OFF_RED);   // [256] reduction scratch

    const int tid     = threadIdx.x;
    const int wave    = tid >> 5;
    const int lane    = tid & 31;
    const int half    = lane >> 4;       // K-half select for A/B frags
    const int l16     = lane & 15;
    const int rowbase = blockIdx.x * ROWS;

    // stage labels / norms into LDS
    for (int i = tid; i < B_N; i += 256) {
        lbl[i]  = labels[i];
        rn[i]   = ws_in[WS_R + i];
        invn[i] = ws_in[WS_INVN + i];
    }
    __syncthreads();

    // ---------------- phase 1: WMMA GEMM tiles + contrastive + distances ----
    const float* arow = F + (size_t)(rowbase + l16) * D_K + 2 * half;
    float cacc = 0.0f;

    for (int t = wave; t < NTILES; t += NWAVES) {
        const int n0 = t * 16;
        const int jcol = n0 + l16;
        const float* brow = F + (size_t)jcol * D_K + 2 * half;

        v8f c = {0.f, 0.f, 0.f, 0.f, 0.f, 0.f, 0.f, 0.f};
#pragma unroll 8
        for (int k = 0; k < D_K; k += 4) {
            v2f a = *(const v2f*)(arow + k);
            v2f b = *(const v2f*)(brow + k);
            // D = A(16x4 f32) x B(4x16 f32) + C, exact fp32 semantics
            c = __builtin_amdgcn_wmma_f32_16x16x4_f32(
                    false, a, false, b, (short)0, c, false, false);
        }

        const float invj = invn[jcol];
        const float rj   = rn[jcol];
        const int   lj   = lbl[jcol];
#pragma unroll
        for (int v = 0; v < 8; ++v) {
            const int   iloc = v + 8 * half;           // C/D VGPR layout
            const int   irow = rowbase + iloc;
            const float g    = c[v];
            const float sim  = g * invn[irow] * invj;
            const bool  same = (lbl[irow] == lj);
            const float pos  = same ? sim : 0.0f;
            cacc += -logf(expf(pos * INV_T) + 1e-8f);  // pos_loss term
            const float negv = same ? 0.0f : sim;
            cacc += fmaxf(C_MARGIN - negv, 0.0f);      // neg_loss term
            float d2 = rn[irow] - 2.0f * g + rj;
            d2 = fmaxf(d2, 0.0f);
            const float dist = (d2 > 0.0f) ? sqrtf(d2) : 0.0f;
            dsm[iloc * DS_STRIDE + jcol] = dist;
        }
    }
    __syncthreads();

    // ---------------- phase 2: triplet scan over stored distance rows -------
    // sum over (p,n): relu(d_ip - d_in + margin) * [lbl p == lbl i][lbl n != lbl i]
    float tacc = 0.0f;
    for (int iloc = 0; iloc < ROWS; ++iloc) {
        const int li = lbl[rowbase + iloc];
        const float* drow = dsm + iloc * DS_STRIDE;
        for (int p = 0; p < B_N; ++p) {             // positives are sparse
            if (lbl[p] == li) {
                const float dp = drow[p] + T_MARGIN;
                for (int n = tid; n < B_N; n += 256) {
                    if (lbl[n] != li) tacc += fmaxf(dp - drow[n], 0.0f);
                }
            }
        }
    }

    // deterministic block reductions -> per-workgroup partials
    const float csum = block_reduce_sum(cacc, red, tid);
    const float tsum = block_reduce_sum(tacc, red, tid);
    if (tid == 0) {
        ws_out[WS_CPART + blockIdx.x] = csum;
        ws_out[WS_TPART + blockIdx.x] = tsum;
    }
}

// -------------------------------------------- focal + label smoothing ------
__global__ void el_focal_ls_kernel(const float* __restrict__ pred,
                                   const int* __restrict__ target,
                                   float* __restrict__ ws) {
    __shared__ float red[256];
    const int tid = threadIdx.x;
    const int row = blockIdx.x;
    const float* pr = pred + (size_t)row * C_CLS;

    float m = -3.4e38f;
    for (int j = tid; j < C_CLS; j += 256) m = fmaxf(m, pr[j]);
    m = block_reduce_max(m, red, tid);

    float se = 0.0f, sp = 0.0f;
    for (int j = tid; j < C_CLS; j += 256) {
        const float p = pr[j];
        se += expf(p - m);
        sp += p;
    }
    se = block_reduce_sum(se, red, tid);
    sp = block_reduce_sum(sp, red, tid);

    if (tid == 0) {
        const float lse = m + logf(se);
        const int   tgt = target[row];
        const float tl  = pr[tgt] - lse;          // target log-prob
        const float ce  = -tl;
        const float pt  = expf(tl);
        const float omp = 1.0f - pt;
        ws[WS_FOCAL + row] = 0.25f * omp * omp * ce;
        const float off = 0.1f / (float)(C_CLS - 1);
        const float sum_logp = sp - (float)C_CLS * lse;
        ws[WS_LSM + row] = -(off * sum_logp + (0.9f - off) * tl);
    }
}

// ------------------------------------------------------------ finalize -----
__global__ void el_finalize_kernel(const float* __restrict__ ws,
                                   float* __restrict__ out) {
    __shared__ float red[256];
    const int tid = threadIdx.x;

    float sf = 0.0f, sl = 0.0f;
    for (int r = tid; r < B_N; r += 256) {
        sf += ws[WS_FOCAL + r];
        sl += ws[WS_LSM + r];
    }
    float sc = 0.0f, st = 0.0f;
    if (tid < 128) {
        sc = ws[WS_CPART + tid];
        st = ws[WS_TPART + tid];
    }
    sc = block_reduce_sum(sc, red, tid);
    st = block_reduce_sum(st, red, tid);
    sf = block_reduce_sum(sf, red, tid);
    sl = block_reduce_sum(sl, red, tid);

    if (tid == 0) {
        const float lc = sc * (1.0f / ((float)B_N * (float)B_N));
        const float lt = st / ((float)B_N + 1e-8f);
        const float lf = sf / (float)B_N;
        const float ls = sl / (float)B_N;
        out[0] = lc;
        out[1] = lt;
        out[2] = lf;
        out[3] = ls;
        out[4] = 0.1f * lc + 0.1f * lt + 0.4f * lf + 0.4f * ls;
    }
}

// ---------------------------------------------------------------------------
extern "C" void kernel_launch(void* const* d_in, const int* in_sizes, int n_in,
                              void* d_out, int out_size, void* d_ws, size_t ws_size,
                              hipStream_t stream) {
    const float* pred     = (const float*)d_in[0];   // [2048,1000]
    const int*   target   = (const int*)d_in[1];     // [2048]
    const float* features = (const float*)d_in[2];   // [2048,512]
    float*       out      = (float*)d_out;           // [5]
    float*       ws       = (float*)d_ws;

    el_norms_kernel<<<B_N / 8, 256, 0, stream>>>(features, ws);
    el_gram_fused_kernel<<<B_N / ROWS, 256, DYN_LDS, stream>>>(features, target, ws, ws);
    el_focal_ls_kernel<<<B_N, 256, 0, stream>>>(pred, target, ws);
    el_finalize_kernel<<<1, 256, 0, stream>>>(ws, out);
}